// GATSecondLayer_11467562680539
// MI455X (gfx1250) — compile-verified
//
#include <hip/hip_runtime.h>
#include <hip/hip_bf16.h>

#define N_NODES 50000
#define N_EDGES 1200000
#define IN_DIM  256
#define OUT_DIM 64
#define SLOPE   0.2f

typedef __attribute__((ext_vector_type(2))) float v2f;
typedef __attribute__((ext_vector_type(8))) float v8f;

// ---------------------------------------------------------------------------
// Kernel 0: init — zero d_out (agg accumulator), init segment-max encoding
// to 0 (== -inf under the monotonic encoding) and denom to 0.
// ---------------------------------------------------------------------------
__global__ void gat_init_kernel(float* __restrict__ out,
                                unsigned* __restrict__ menc,
                                float* __restrict__ denom) {
    int i = blockIdx.x * blockDim.x + threadIdx.x;
    if (i < N_NODES * OUT_DIM) out[i] = 0.0f;
    if (i < N_NODES) { menc[i] = 0u; denom[i] = 0.0f; }
}

// ---------------------------------------------------------------------------
// Kernel 1: z = h @ W_fc^T via V_WMMA_F32_16X16X4_F32.
// One wave computes a 16(M) x 64(N) slab: 4 accumulators of 16x16.
// A (16x4 f32): lane l<16 -> row M=l, VGPR0/1 = K {0,1}; lanes 16-31 -> K {2,3}.
// B (4x16 f32): lane l<16 -> col N=l, VGPR0/1 = K {0,1}; lanes 16-31 -> K {2,3}.
// C (16x16 f32): VGPR i -> M = i + 8*(lane>=16), N = lane&15.
// ---------------------------------------------------------------------------
__global__ void __launch_bounds__(256)
gat_gemm_z_kernel(const float* __restrict__ h,
                  const float* __restrict__ Wfc,   // (OUT_DIM, IN_DIM) row-major
                  float* __restrict__ z) {         // (N_NODES, OUT_DIM)
    const int wave  = threadIdx.x >> 5;
    const int lane  = threadIdx.x & 31;
    const int mtile = blockIdx.x * (blockDim.x >> 5) + wave;
    if (mtile * 16 >= N_NODES) return;             // wave-uniform: EXEC stays all-1s

    const int row0 = mtile * 16;
    const int lr   = lane & 15;                    // row (A) / col (B) within tile
    const int kg   = (lane >> 4) << 1;             // K sub-offset: 0 or 2

    v8f acc0 = {}, acc1 = {}, acc2 = {}, acc3 = {};
    const float* hrow = h + (size_t)(row0 + lr) * IN_DIM;
    const float* w0 = Wfc + (size_t)(0 * 16 + lr) * IN_DIM;
    const float* w1 = Wfc + (size_t)(1 * 16 + lr) * IN_DIM;
    const float* w2 = Wfc + (size_t)(2 * 16 + lr) * IN_DIM;
    const float* w3 = Wfc + (size_t)(3 * 16 + lr) * IN_DIM;

    for (int k0 = 0; k0 < IN_DIM; k0 += 4) {
        const int ko = k0 + kg;
        v2f a  = *(const v2f*)(hrow + ko);
        v2f b0 = *(const v2f*)(w0 + ko);
        v2f b1 = *(const v2f*)(w1 + ko);
        v2f b2 = *(const v2f*)(w2 + ko);
        v2f b3 = *(const v2f*)(w3 + ko);
        acc0 = __builtin_amdgcn_wmma_f32_16x16x4_f32(false, a, false, b0, (short)0, acc0, false, false);
        acc1 = __builtin_amdgcn_wmma_f32_16x16x4_f32(false, a, false, b1, (short)0, acc1, false, false);
        acc2 = __builtin_amdgcn_wmma_f32_16x16x4_f32(false, a, false, b2, (short)0, acc2, false, false);
        acc3 = __builtin_amdgcn_wmma_f32_16x16x4_f32(false, a, false, b3, (short)0, acc3, false, false);
    }

    const int rbase = row0 + ((lane >> 4) << 3);
#pragma unroll
    for (int i = 0; i < 8; ++i) {
        float* zr = z + (size_t)(rbase + i) * OUT_DIM + lr;
        zr[0]  = acc0[i];
        zr[16] = acc1[i];
        zr[32] = acc2[i];
        zr[48] = acc3[i];
    }
}

// ---------------------------------------------------------------------------
// Kernel 2: s_src = z @ a[:64], s_dst = z @ a[64:]
// ---------------------------------------------------------------------------
__global__ void gat_score_kernel(const float* __restrict__ z,
                                 const float* __restrict__ Wattn,   // (1, 128)
                                 float* __restrict__ s_src,
                                 float* __restrict__ s_dst) {
    int n = blockIdx.x * blockDim.x + threadIdx.x;
    if (n >= N_NODES) return;
    const float* zr = z + (size_t)n * OUT_DIM;
    float ss = 0.0f, sd = 0.0f;
#pragma unroll 8
    for (int j = 0; j < OUT_DIM; ++j) {
        float v = zr[j];
        ss += v * Wattn[j];
        sd += v * Wattn[OUT_DIM + j];
    }
    s_src[n] = ss;
    s_dst[n] = sd;
}

// Monotonic unsigned encoding of float (order-preserving) for atomic max.
__device__ __forceinline__ unsigned fenc(float f) {
    unsigned u = __float_as_uint(f);
    return (u & 0x80000000u) ? ~u : (u | 0x80000000u);
}
__device__ __forceinline__ float fdec(unsigned u) {
    unsigned b = (u & 0x80000000u) ? (u & 0x7FFFFFFFu) : ~u;
    return __uint_as_float(b);
}

// ---------------------------------------------------------------------------
// Kernel 3: e = leaky_relu(s_src[src] + s_dst[dst]); segment max via
// global_atomic_max_u32 on the monotonic encoding.
// ---------------------------------------------------------------------------
__global__ void gat_edge_max_kernel(const int* __restrict__ src,
                                    const int* __restrict__ dst,
                                    const float* __restrict__ s_src,
                                    const float* __restrict__ s_dst,
                                    float* __restrict__ ework,
                                    unsigned* __restrict__ menc) {
    int i = blockIdx.x * blockDim.x + threadIdx.x;
    if (i >= N_EDGES) return;
    float e = s_src[src[i]] + s_dst[dst[i]];
    e = (e > 0.0f) ? e : SLOPE * e;
    ework[i] = e;
    atomicMax(&menc[dst[i]], fenc(e));
}

// ---------------------------------------------------------------------------
// Kernel 4: ex = exp(e - m[dst]); denom[dst] += ex (atomic f32 add).
// ---------------------------------------------------------------------------
__global__ void gat_edge_exp_kernel(const int* __restrict__ dst,
                                    const unsigned* __restrict__ menc,
                                    float* __restrict__ ework,
                                    float* __restrict__ denom) {
    int i = blockIdx.x * blockDim.x + threadIdx.x;
    if (i >= N_EDGES) return;
    int d = dst[i];
    float ex = expf(ework[i] - fdec(menc[d]));
    ework[i] = ex;
    atomicAdd(&denom[d], ex);
}

// ---------------------------------------------------------------------------
// Kernel 5: agg[dst] += (ex/denom[dst]) * z[src], one (edge, dim-pair of 1)
// per thread; 32 lanes cover 32 contiguous dims -> coalesced z reads, L2
// resident gather (z is 12.8 MB << 192 MB L2).
// ---------------------------------------------------------------------------
__global__ void gat_edge_agg_kernel(const int* __restrict__ src,
                                    const int* __restrict__ dst,
                                    const float* __restrict__ ework,
                                    const float* __restrict__ denom,
                                    const float* __restrict__ z,
                                    float* __restrict__ agg) {
    long long t = (long long)blockIdx.x * blockDim.x + threadIdx.x;
    int edge = (int)(t >> 6);
    int d    = (int)(t & 63);
    if (edge >= N_EDGES) return;
    int s  = src[edge];
    int dd = dst[edge];
    float alpha = ework[edge] / denom[dd];
    atomicAdd(&agg[(size_t)dd * OUT_DIM + d], alpha * z[(size_t)s * OUT_DIM + d]);
}

// ---------------------------------------------------------------------------
// Kernel 6: out = elu(agg) in place.
// ---------------------------------------------------------------------------
__global__ void gat_elu_kernel(float* __restrict__ out) {
    int i = blockIdx.x * blockDim.x + threadIdx.x;
    if (i >= N_NODES * OUT_DIM) return;
    float x = out[i];
    out[i] = (x > 0.0f) ? x : (expf(x) - 1.0f);
}

extern "C" void kernel_launch(void* const* d_in, const int* in_sizes, int n_in,
                              void* d_out, int out_size, void* d_ws, size_t ws_size,
                              hipStream_t stream) {
    const float* h     = (const float*)d_in[0];
    const float* Wfc   = (const float*)d_in[1];
    const float* Wattn = (const float*)d_in[2];
    const int*   src   = (const int*)d_in[3];
    const int*   dst   = (const int*)d_in[4];
    float* out = (float*)d_out;

    // Workspace layout (floats): z | s_src | s_dst | menc | denom | ework
    float*    z     = (float*)d_ws;
    float*    s_src = z + (size_t)N_NODES * OUT_DIM;
    float*    s_dst = s_src + N_NODES;
    unsigned* menc  = (unsigned*)(s_dst + N_NODES);
    float*    denom = (float*)(menc + N_NODES);
    float*    ework = denom + N_NODES;

    const int B = 256;

    // 0) init accumulators
    gat_init_kernel<<<(N_NODES * OUT_DIM + B - 1) / B, B, 0, stream>>>(out, menc, denom);

    // 1) z = h @ Wfc^T   (WMMA: 3125 M-tiles, 8 waves/block)
    {
        int waves_needed = N_NODES / 16;              // 3125
        int blocks = (waves_needed + 7) / 8;          // 391
        gat_gemm_z_kernel<<<blocks, B, 0, stream>>>(h, Wfc, z);
    }

    // 2) attention scores
    gat_score_kernel<<<(N_NODES + B - 1) / B, B, 0, stream>>>(z, Wattn, s_src, s_dst);

    // 3) e + segment max
    gat_edge_max_kernel<<<(N_EDGES + B - 1) / B, B, 0, stream>>>(src, dst, s_src, s_dst, ework, menc);

    // 4) exp + segment sum
    gat_edge_exp_kernel<<<(N_EDGES + B - 1) / B, B, 0, stream>>>(dst, menc, ework, denom);

    // 5) weighted aggregation (64 threads per edge)
    {
        long long total = (long long)N_EDGES * OUT_DIM;
        int blocks = (int)((total + B - 1) / B);
        gat_edge_agg_kernel<<<blocks, B, 0, stream>>>(src, dst, ework, denom, z, out);
    }

    // 6) elu
    gat_elu_kernel<<<(N_NODES * OUT_DIM + B - 1) / B, B, 0, stream>>>(out);
}